// GATResBlock_10222022164891
// MI455X (gfx1250) — compile-verified
//
#include <hip/hip_runtime.h>
#include <math.h>

#define N_NODES   50000
#define IN_F      128
#define H_HEADS   4
#define C_CH      32
#define HC        128           // H*C
#define E_EDGES   800000
#define NEG_SLOPE 0.2f

typedef __attribute__((ext_vector_type(2))) float v2f;
typedef __attribute__((ext_vector_type(8))) float v8f;

// ---------- monotonic float <-> uint mapping for atomic max ----------
__device__ __forceinline__ unsigned fmap(float f) {
    unsigned u = __float_as_uint(f);
    return (u & 0x80000000u) ? ~u : (u | 0x80000000u);
}
__device__ __forceinline__ float funmap(unsigned u) {
    unsigned v = (u & 0x80000000u) ? (u & 0x7FFFFFFFu) : ~u;
    return __uint_as_float(v);
}

// ---------- fused dual GEMM: xp = x@W^T, xsk = x@skipW^T (fp32 WMMA) ----------
// grid.x = N/16 row tiles, 512 threads = 16 waves.
// waves 0..7  -> W      column tiles 0..7  -> xp
// waves 8..15 -> skip_W column tiles 0..7  -> xsk
__global__ __launch_bounds__(512)
void gat_gemm_dual(const float* __restrict__ x,
                   const float* __restrict__ W,
                   const float* __restrict__ skipW,
                   float* __restrict__ xp,
                   float* __restrict__ xsk)
{
    const int lane     = threadIdx.x & 31;
    const int wave     = threadIdx.x >> 5;      // 0..15
    const int whichW   = wave >> 3;             // 0 -> W, 1 -> skipW
    const int col_base = (wave & 7) * 16;
    const int row_base = blockIdx.x * 16;

    const float* Wm  = whichW ? skipW : W;
    float*       dst = whichW ? xsk   : xp;

    // A (16x4 fp32): lanes 0-15 = M rows, K pair {0,1}; lanes 16-31 same rows, K pair {2,3}
    // B (4x16 fp32): lanes 0-15 = N cols, K pair {0,1}; lanes 16-31 same cols, K pair {2,3}
    const int idx15  = lane & 15;
    const int khalf  = (lane >> 4) << 1;        // 0 or 2

    const float* arow = x  + (size_t)(row_base + idx15) * IN_F + khalf;
    const float* brow = Wm + (size_t)(col_base + idx15) * IN_F + khalf;

    v8f acc = {};
#pragma unroll
    for (int k0 = 0; k0 < IN_F; k0 += 4) {
        v2f a = *(const v2f*)(arow + k0);
        v2f b = *(const v2f*)(brow + k0);
        acc = __builtin_amdgcn_wmma_f32_16x16x4_f32(
            /*neg_a=*/false, a, /*neg_b=*/false, b,
            /*c_mod=*/(short)0, acc, /*reuse_a=*/false, /*reuse_b=*/false);
    }

    // C/D layout: VGPR r -> M = r + 8*(lane>=16), N = lane&15
    const int nCol = lane & 15;
    const int mOff = (lane >> 4) << 3;          // 0 or 8
#pragma unroll
    for (int r = 0; r < 8; ++r) {
        dst[(size_t)(row_base + mOff + r) * HC + col_base + nCol] = acc[r];
    }
}

// ---------- init: zero d_out accumulator, denom; reset mmax ----------
__global__ void gat_init(float* __restrict__ out_acc,
                         unsigned* __restrict__ mmax,
                         float* __restrict__ denom)
{
    int t = blockIdx.x * blockDim.x + threadIdx.x;
    if (t < N_NODES * HC) out_acc[t] = 0.0f;
    if (t < N_NODES * H_HEADS) { mmax[t] = 0u; denom[t] = 0.0f; }
}

// ---------- per-node attention logits: a_src/a_dst [N,H] ----------
__global__ void gat_logits(const float* __restrict__ xp,
                           const float* __restrict__ att_src,
                           const float* __restrict__ att_dst,
                           float* __restrict__ a_src,
                           float* __restrict__ a_dst)
{
    int t = blockIdx.x * blockDim.x + threadIdx.x;
    if (t >= N_NODES * H_HEADS) return;
    int h = t & (H_HEADS - 1);
    const float* row = xp + (size_t)(t >> 2) * HC + h * C_CH;
    const float* as  = att_src + h * C_CH;
    const float* ad  = att_dst + h * C_CH;
    float s = 0.0f, d = 0.0f;
#pragma unroll
    for (int c = 0; c < C_CH; ++c) {
        float v = row[c];
        s += v * as[c];
        d += v * ad[c];
    }
    a_src[t] = s;
    a_dst[t] = d;
}

// ---------- edge logits + LeakyReLU + segment max ----------
__global__ void gat_edge_max(const long long* __restrict__ src,
                             const long long* __restrict__ dstp,
                             const float* __restrict__ a_src,
                             const float* __restrict__ a_dst,
                             float* __restrict__ ebuf,
                             unsigned* __restrict__ mmax)
{
    int t = blockIdx.x * blockDim.x + threadIdx.x;
    if (t >= E_EDGES * H_HEADS) return;
    int e = t >> 2;
    int h = t & (H_HEADS - 1);
    size_t s = (size_t)src[e];
    size_t d = (size_t)dstp[e];
    float v = a_src[s * H_HEADS + h] + a_dst[d * H_HEADS + h];
    v = (v > 0.0f) ? v : NEG_SLOPE * v;
    ebuf[t] = v;
    atomicMax(&mmax[d * H_HEADS + h], fmap(v));
}

// ---------- exp(e - m[dst]) + segment sum ----------
__global__ void gat_edge_exp(const long long* __restrict__ dstp,
                             const unsigned* __restrict__ mmax,
                             float* __restrict__ ebuf,       // in: e, out: ex
                             float* __restrict__ denom)
{
    int t = blockIdx.x * blockDim.x + threadIdx.x;
    if (t >= E_EDGES * H_HEADS) return;
    int e = t >> 2;
    int h = t & (H_HEADS - 1);
    size_t d = (size_t)dstp[e];
    float m  = funmap(mmax[d * H_HEADS + h]);
    float ex = expf(ebuf[t] - m);
    ebuf[t] = ex;
    atomicAdd(&denom[d * H_HEADS + h], ex);
}

// ---------- weighted message scatter-add into out accumulator ----------
// one thread per (edge, hc): coalesced xp gather + coalesced atomics
__global__ void gat_edge_agg(const long long* __restrict__ src,
                             const long long* __restrict__ dstp,
                             const float* __restrict__ ebuf,  // ex
                             const float* __restrict__ denom,
                             const float* __restrict__ xp,
                             float* __restrict__ out_acc)
{
    long long t = (long long)blockIdx.x * blockDim.x + threadIdx.x;
    if (t >= (long long)E_EDGES * HC) return;
    int e  = (int)(t >> 7);
    int hc = (int)(t & (HC - 1));
    int h  = hc >> 5;
    size_t s = (size_t)src[e];
    size_t d = (size_t)dstp[e];
    float alpha = ebuf[(size_t)e * H_HEADS + h] /
                  (denom[d * H_HEADS + h] + 1e-16f);
    float val = alpha * xp[s * HC + hc];
    atomicAdd(&out_acc[d * HC + hc], val);
}

// ---------- finalize: out = elu(agg + bias + xsk + skip_b) in place ----------
__global__ void gat_final(float* __restrict__ out,
                          const float* __restrict__ bias,
                          const float* __restrict__ xsk,
                          const float* __restrict__ skip_b)
{
    int t = blockIdx.x * blockDim.x + threadIdx.x;
    if (t >= N_NODES * HC) return;
    int hc = t & (HC - 1);
    float v = out[t] + bias[hc] + xsk[t] + skip_b[hc];
    out[t] = (v > 0.0f) ? v : expm1f(v);
}

extern "C" void kernel_launch(void* const* d_in, const int* in_sizes, int n_in,
                              void* d_out, int out_size, void* d_ws, size_t ws_size,
                              hipStream_t stream)
{
    const float*     x        = (const float*)d_in[0];
    const long long* eidx     = (const long long*)d_in[1];   // int64 [2, E]
    const float*     W        = (const float*)d_in[2];
    const float*     att_src  = (const float*)d_in[3];
    const float*     att_dst  = (const float*)d_in[4];
    const float*     bias     = (const float*)d_in[5];
    const float*     skip_W   = (const float*)d_in[6];
    const float*     skip_b   = (const float*)d_in[7];
    float*           out      = (float*)d_out;

    const long long* src  = eidx;            // row 0
    const long long* dstp = eidx + E_EDGES;  // row 1

    // workspace layout (floats)
    float* ws     = (float*)d_ws;
    float* xp     = ws;                           // N*HC
    float* xsk    = xp   + (size_t)N_NODES * HC;  // N*HC
    float* a_src  = xsk  + (size_t)N_NODES * HC;  // N*H
    float* a_dst  = a_src + (size_t)N_NODES * H_HEADS;
    unsigned* mmax = (unsigned*)(a_dst + (size_t)N_NODES * H_HEADS); // N*H (u32)
    float* denom  = (float*)(mmax + (size_t)N_NODES * H_HEADS);      // N*H
    float* ebuf   = denom + (size_t)N_NODES * H_HEADS;               // E*H

    // 1) fused dual GEMM (fp32 WMMA): xp, xsk
    gat_gemm_dual<<<N_NODES / 16, 512, 0, stream>>>(x, W, skip_W, xp, xsk);

    // 2) init accumulators
    {
        int n = N_NODES * HC;
        gat_init<<<(n + 255) / 256, 256, 0, stream>>>(out, mmax, denom);
    }

    // 3) per-node attention logits
    {
        int n = N_NODES * H_HEADS;
        gat_logits<<<(n + 255) / 256, 256, 0, stream>>>(xp, att_src, att_dst, a_src, a_dst);
    }

    // 4) edge logits + segment max
    {
        int n = E_EDGES * H_HEADS;
        gat_edge_max<<<(n + 255) / 256, 256, 0, stream>>>(src, dstp, a_src, a_dst, ebuf, mmax);
    }

    // 5) exp + segment sum
    {
        int n = E_EDGES * H_HEADS;
        gat_edge_exp<<<(n + 255) / 256, 256, 0, stream>>>(dstp, mmax, ebuf, denom);
    }

    // 6) weighted scatter-add aggregation
    {
        long long n = (long long)E_EDGES * HC;
        int blocks = (int)((n + 255) / 256);
        gat_edge_agg<<<blocks, 256, 0, stream>>>(src, dstp, ebuf, denom, xp, out);
    }

    // 7) bias + skip + ELU
    {
        int n = N_NODES * HC;
        gat_final<<<(n + 255) / 256, 256, 0, stream>>>(out, bias, xsk, skip_b);
    }
}